// DGL_GMM_39599598469539
// MI455X (gfx1250) — compile-verified
//
#include <hip/hip_runtime.h>
#include <math.h>

typedef __attribute__((ext_vector_type(2))) float v2f;
typedef __attribute__((ext_vector_type(8))) float v8f;

#define N_NODES   50000
#define N_EDGES   800000
#define IN_FEATS  128
#define N_HIDDEN  64
#define OUT_FEATS 40
#define OUT_PAD   48
#define DIM       8

// ---------------------------------------------------------------- utilities
__global__ __launch_bounds__(256) void zero_f32(float* __restrict__ p, int n) {
  int i = blockIdx.x * blockDim.x + threadIdx.x;
  int stride = gridDim.x * blockDim.x;
  for (; i < n; i += stride) p[i] = 0.0f;
}

// Pad W2 [64 x 40] -> [64 x 48] with zero columns so the WMMA B-load is
// unconditional (EXEC must be all 1s at every v_wmma).
__global__ __launch_bounds__(256) void pad_w2(const float* __restrict__ W2,
                                              float* __restrict__ W2p) {
  int i = blockIdx.x * blockDim.x + threadIdx.x;
  if (i >= N_HIDDEN * OUT_PAD) return;
  int k = i / OUT_PAD, c = i % OUT_PAD;
  W2p[i] = (c < OUT_FEATS) ? W2[k * OUT_FEATS + c] : 0.0f;
}

// -------------------------------------------- per-edge gaussian weights (K=1)
__global__ __launch_bounds__(256) void gw_kernel(
    const float* __restrict__ pseudo,
    const float* __restrict__ mu1, const float* __restrict__ is1,
    const float* __restrict__ mu2, const float* __restrict__ is2,
    float* __restrict__ gw1, float* __restrict__ gw2) {
  int e = blockIdx.x * blockDim.x + threadIdx.x;
  if (e >= N_EDGES) return;
  float s1 = 0.0f, s2 = 0.0f;
#pragma unroll
  for (int d = 0; d < DIM; ++d) {
    float p  = pseudo[e * DIM + d];
    float d1 = p - mu1[d], w1 = is1[d];
    float d2 = p - mu2[d], w2 = is2[d];
    s1 += d1 * d1 * w1 * w1;
    s2 += d2 * d2 * w2 * w2;
  }
  gw1[e] = expf(-0.5f * s1);
  gw2[e] = expf(-0.5f * s2);
}

// ------------------------------------------------- GEMM1: H1 = X @ W1 (WMMA)
// One wave computes one 16x16 output tile; 4 waves cover the 64 hidden cols.
__global__ __launch_bounds__(128) void gemm1_wmma(
    const float* __restrict__ X,    // [N, 128]
    const float* __restrict__ W,    // [128, 64]
    float* __restrict__ H) {        // [N, 64]
  const int wave = threadIdx.x >> 5;
  const int lane = threadIdx.x & 31;
  const int m    = lane & 15;
  const int hi   = lane >> 4;           // 0 or 1 -> K offset 0 / 2
  const int r0   = blockIdx.x * 16;
  const int c0   = wave * 16;
  const float* __restrict__ arow = X + (size_t)(r0 + m) * IN_FEATS;

  v8f c = {};
  for (int kk = 0; kk < IN_FEATS; kk += 4) {
    const int ka = kk + 2 * hi;
    v2f a, b;
    a.x = arow[ka];
    a.y = arow[ka + 1];
    b.x = W[(size_t)ka * N_HIDDEN + c0 + m];
    b.y = W[(size_t)(ka + 1) * N_HIDDEN + c0 + m];
    c = __builtin_amdgcn_wmma_f32_16x16x4_f32(false, a, false, b,
                                              (short)0, c, false, false);
  }
#pragma unroll
  for (int i = 0; i < 8; ++i)
    H[(size_t)(r0 + i + 8 * hi) * N_HIDDEN + c0 + m] = c[i];
}

// --------------------------------- GEMM2: H2 = (agg1 + b1) @ W2pad  (WMMA)
__global__ __launch_bounds__(96) void gemm2_wmma(
    const float* __restrict__ X,     // [N, 64] (agg1)
    const float* __restrict__ bias,  // [64] (b1, folded into A)
    const float* __restrict__ W,     // [64, 48] padded
    float* __restrict__ H) {         // [N, 48]
  const int wave = threadIdx.x >> 5;    // 0..2 -> 48 output cols
  const int lane = threadIdx.x & 31;
  const int m    = lane & 15;
  const int hi   = lane >> 4;
  const int r0   = blockIdx.x * 16;
  const int c0   = wave * 16;
  const float* __restrict__ arow = X + (size_t)(r0 + m) * N_HIDDEN;

  v8f c = {};
  for (int kk = 0; kk < N_HIDDEN; kk += 4) {
    const int ka = kk + 2 * hi;
    v2f a, b;
    a.x = arow[ka]     + bias[ka];
    a.y = arow[ka + 1] + bias[ka + 1];
    b.x = W[(size_t)ka * OUT_PAD + c0 + m];
    b.y = W[(size_t)(ka + 1) * OUT_PAD + c0 + m];
    c = __builtin_amdgcn_wmma_f32_16x16x4_f32(false, a, false, b,
                                              (short)0, c, false, false);
  }
#pragma unroll
  for (int i = 0; i < 8; ++i)
    H[(size_t)(r0 + i + 8 * hi) * OUT_PAD + c0 + m] = c[i];
}

// ----------------------------- edge gather/scale/scatter, 64 feats (layer 1)
// One wave per edge: lanes cover feats [0,32) and [32,64) -> two coalesced
// 128B lines per access, atomic f32 adds resolve in L2.
__global__ __launch_bounds__(256) void scatter64(
    const float* __restrict__ H, const int* __restrict__ src,
    const int* __restrict__ dst, const float* __restrict__ gw,
    float* __restrict__ agg) {
  int e    = (blockIdx.x * blockDim.x + threadIdx.x) >> 5;
  int lane = threadIdx.x & 31;
  if (e >= N_EDGES) return;
  int   s = src[e], d = dst[e];
  float g = gw[e];
  float v0 = H[(size_t)s * N_HIDDEN + lane] * g;
  float v1 = H[(size_t)s * N_HIDDEN + lane + 32] * g;
  atomicAdd(&agg[(size_t)d * N_HIDDEN + lane], v0);
  atomicAdd(&agg[(size_t)d * N_HIDDEN + lane + 32], v1);
}

// ----------------------------- edge gather/scale/scatter, 40 feats (layer 2)
__global__ __launch_bounds__(256) void scatter40(
    const float* __restrict__ H, const int* __restrict__ src,
    const int* __restrict__ dst, const float* __restrict__ gw,
    float* __restrict__ agg) {
  int e    = (blockIdx.x * blockDim.x + threadIdx.x) >> 5;
  int lane = threadIdx.x & 31;
  if (e >= N_EDGES) return;
  int   s = src[e], d = dst[e];
  float g = gw[e];
  float v0 = H[(size_t)s * OUT_PAD + lane] * g;
  atomicAdd(&agg[(size_t)d * OUT_FEATS + lane], v0);
  if (lane < OUT_FEATS - 32) {
    float v1 = H[(size_t)s * OUT_PAD + lane + 32] * g;
    atomicAdd(&agg[(size_t)d * OUT_FEATS + lane + 32], v1);
  }
}

// --------------------------------- row-wise log_softmax over 40 classes
// One wave per row; lane holds feats {lane, lane+32}; wave32 shfl reductions.
__global__ __launch_bounds__(256) void logsoftmax40(
    const float* __restrict__ agg, const float* __restrict__ b,
    float* __restrict__ out) {
  int r    = (blockIdx.x * blockDim.x + threadIdx.x) >> 5;
  int lane = threadIdx.x & 31;
  if (r >= N_NODES) return;
  const float NEG = -3.402823466e+38f;
  float x0 = agg[(size_t)r * OUT_FEATS + lane] + b[lane];
  bool  hasHi = (lane + 32) < OUT_FEATS;
  float x1 = hasHi ? (agg[(size_t)r * OUT_FEATS + lane + 32] + b[lane + 32]) : NEG;

  float mx = fmaxf(x0, x1);
#pragma unroll
  for (int off = 16; off > 0; off >>= 1)
    mx = fmaxf(mx, __shfl_xor(mx, off, 32));

  float s = expf(x0 - mx) + (hasHi ? expf(x1 - mx) : 0.0f);
#pragma unroll
  for (int off = 16; off > 0; off >>= 1)
    s += __shfl_xor(s, off, 32);

  float lse = mx + logf(s);
  out[(size_t)r * OUT_FEATS + lane] = x0 - lse;
  if (hasHi) out[(size_t)r * OUT_FEATS + lane + 32] = x1 - lse;
}

// --------------------------------------------------------------------------
extern "C" void kernel_launch(void* const* d_in, const int* in_sizes, int n_in,
                              void* d_out, int out_size, void* d_ws, size_t ws_size,
                              hipStream_t stream) {
  const float* features = (const float*)d_in[0];
  const float* pseudo   = (const float*)d_in[1];
  const int*   src      = (const int*)d_in[2];
  const int*   dst      = (const int*)d_in[3];
  const float* W1       = (const float*)d_in[4];
  const float* b1       = (const float*)d_in[5];
  const float* mu1      = (const float*)d_in[6];
  const float* is1      = (const float*)d_in[7];
  const float* W2       = (const float*)d_in[8];
  const float* b2       = (const float*)d_in[9];
  const float* mu2      = (const float*)d_in[10];
  const float* is2      = (const float*)d_in[11];

  float* ws   = (float*)d_ws;
  float* h1   = ws;                                   // 50000*64
  float* agg1 = h1   + (size_t)N_NODES * N_HIDDEN;    // 50000*64
  float* h2   = agg1 + (size_t)N_NODES * N_HIDDEN;    // 50000*48
  float* agg2 = h2   + (size_t)N_NODES * OUT_PAD;     // 50000*40
  float* gw1  = agg2 + (size_t)N_NODES * OUT_FEATS;   // 800000
  float* gw2  = gw1  + (size_t)N_EDGES;               // 800000
  float* w2p  = gw2  + (size_t)N_EDGES;               // 64*48
  float* out  = (float*)d_out;

  // accumulators must be zeroed every call (ws is poisoned, not re-zeroed)
  zero_f32<<<2048, 256, 0, stream>>>(agg1, N_NODES * N_HIDDEN);
  zero_f32<<<2048, 256, 0, stream>>>(agg2, N_NODES * OUT_FEATS);
  pad_w2<<<(N_HIDDEN * OUT_PAD + 255) / 256, 256, 0, stream>>>(W2, w2p);
  gw_kernel<<<(N_EDGES + 255) / 256, 256, 0, stream>>>(pseudo, mu1, is1, mu2, is2, gw1, gw2);

  gemm1_wmma<<<N_NODES / 16, 128, 0, stream>>>(features, W1, h1);
  scatter64<<<N_EDGES / 8, 256, 0, stream>>>(h1, src, dst, gw1, agg1);

  gemm2_wmma<<<N_NODES / 16, 96, 0, stream>>>(agg1, b1, w2p, h2);
  scatter40<<<N_EDGES / 8, 256, 0, stream>>>(h2, src, dst, gw2, agg2);

  logsoftmax40<<<N_NODES / 8, 256, 0, stream>>>(agg2, b2, out);
}